// IFConv2D_79894981640820
// MI455X (gfx1250) — compile-verified
//
#include <hip/hip_runtime.h>
#include <stdint.h>

typedef __attribute__((ext_vector_type(2))) float v2f;
typedef __attribute__((ext_vector_type(8))) float v8f;

#define C_OUT   64
#define K_FEAT  288
#define NSORT   512
#define N_TILES 18              // 18 * 16 == 288
#define EXPMAX  54.598150033144236f

// ---------------------------------------------------------------------------
// Kernel 1: transpose weight [C_OUT][K_FEAT] -> wT [K_FEAT][C_OUT] so that the
// per-sorted-position gather wT[idx*64 + c] is lane-coalesced (16 consecutive
// floats per half-wave -> 1 cacheline).
// ---------------------------------------------------------------------------
__global__ void ttfs_weight_transpose(const float* __restrict__ w,
                                      float* __restrict__ wT) {
    int e = blockIdx.x * blockDim.x + threadIdx.x;
    if (e < C_OUT * K_FEAT) {
        int c = e & (C_OUT - 1);
        int k = e >> 6;
        wT[e] = w[c * K_FEAT + k];
    }
}

// ---------------------------------------------------------------------------
// Kernel 2: one block (128 thr = 4 wave32) per output pixel.
//   phase 1: build 288 (value||index) keys, pad to 512, bitonic sort in LDS
//   phase 2: each wave owns a 16-neuron column block; prefix sums over the
//            sorted order computed as blocked triangular matmuls with
//            v_wmma_f32_16x16x4_f32 (carry broadcast enters via the C operand)
//   phase 3: division-free spike-time selection + min reduce, one divide/out
// ---------------------------------------------------------------------------
__global__ __launch_bounds__(128)
void ttfs_conv_kernel(const float* __restrict__ x,
                      const float* __restrict__ wT,
                      float* __restrict__ out) {
    __shared__ uint64_t keys[NSORT];
    __shared__ float    slds[K_FEAT + 16];
    __shared__ int      ilds[K_FEAT];

    const int tid = threadIdx.x;
    const int p   = blockIdx.x;          // pixel id
    const int b   = p >> 10;
    const int l   = p & 1023;
    const int ho  = l >> 5;
    const int wo  = l & 31;

    // ---- phase 1: unfold (3x3, pad 1) + threshold + pack sort keys --------
    for (int j = tid; j < NSORT; j += 128) {
        if (j < K_FEAT) {
            int ci = j / 9;
            int r  = j - ci * 9;
            int kh = r / 3;
            int kw = r - kh * 3;
            int h  = ho + kh - 1;
            int w  = wo + kw - 1;
            float v = EXPMAX;                    // zero-pad -> <0.1 -> exp(4)
            if ((unsigned)h < 32u && (unsigned)w < 32u) {
                v = x[((b * 32 + ci) * 32 + h) * 32 + w];
                if (v < 0.1f) v = EXPMAX;
            }
            keys[j] = ((uint64_t)__float_as_uint(v) << 32) | (uint32_t)j;
        } else {
            keys[j] = ((uint64_t)0x7F7FFFFFu << 32) | (uint32_t)j; // FLT_MAX pad
        }
    }
    __syncthreads();

    // ---- bitonic sort, 512 ascending (positive floats sort as uints) ------
    for (int k = 2; k <= NSORT; k <<= 1) {
        for (int jj = k >> 1; jj > 0; jj >>= 1) {
            for (int i = tid; i < NSORT; i += 128) {
                int partner = i ^ jj;
                if (partner > i) {
                    uint64_t a = keys[i], c = keys[partner];
                    bool asc = ((i & k) == 0);
                    if ((a > c) == asc) { keys[i] = c; keys[partner] = a; }
                }
            }
            __syncthreads();
        }
    }

    for (int j = tid; j < K_FEAT; j += 128) {
        uint64_t kv = keys[j];
        slds[j] = __uint_as_float((uint32_t)(kv >> 32));
        ilds[j] = (int)(kv & 0x1FF);
    }
    if (tid == 0) slds[K_FEAT] = 1.0f;   // next-spike pad = exp(0)
    __syncthreads();

    // ---- phase 2: WMMA triangular scan ------------------------------------
    const int lane  = tid & 31;
    const int wave  = tid >> 5;          // column block 0..3 (16 neurons each)
    const int n     = lane & 15;         // N coordinate
    const int half  = lane >> 4;         // 0: rows 0..7 / K {0,1}; 1: rows 8..15 / K {2,3}
    const int cbase = wave * 16;

    // A chunks: 16x4 lower-triangular ones. f32 A layout (16x16x4):
    //   lane m = lane&15; vgpr0 = K (half?2:0)+kbase, vgpr1 = K (half?3:1)+kbase
    v2f A[4];
#pragma unroll
    for (int kc = 0; kc < 4; ++kc) {
        int k0 = kc * 4 + (half ? 2 : 0);
        A[kc].x = (k0     <= n) ? 1.0f : 0.0f;
        A[kc].y = (k0 + 1 <= n) ? 1.0f : 0.0f;
    }

    float carry_w = 0.0f, carry_wi = 0.0f;
    float best_n = EXPMAX, best_d = 1.0f;    // spike time tracked as fraction

    for (int t = 0; t < N_TILES; ++t) {
        const int jbase = t * 16;
        v8f cw, cwi;
#pragma unroll
        for (int r = 0; r < 8; ++r) { cw[r] = carry_w; cwi[r] = carry_wi; }

#pragma unroll
        for (int kc = 0; kc < 4; ++kc) {
            int kr0 = kc * 4 + (half ? 2 : 0);
            int j0  = jbase + kr0;
            int j1  = j0 + 1;
            int i0  = ilds[j0];
            int i1  = ilds[j1];
            float s0 = slds[j0];
            float s1 = slds[j1];
            float w0 = wT[i0 * C_OUT + cbase + n];   // coalesced 16-lane gather
            float w1 = wT[i1 * C_OUT + cbase + n];
            v2f Bw;  Bw.x  = w0;      Bw.y  = w1;
            v2f Bwi; Bwi.x = w0 * s0; Bwi.y = w1 * s1;
            cw  = __builtin_amdgcn_wmma_f32_16x16x4_f32(false, A[kc], false, Bw,
                                                        (short)0, cw,  false, false);
            cwi = __builtin_amdgcn_wmma_f32_16x16x4_f32(false, A[kc], false, Bwi,
                                                        (short)0, cwi, false, false);
        }

        // ---- phase 3 epilogue: D layout VGPR r -> M = r + 8*half, N = n ----
#pragma unroll
        for (int r = 0; r < 8; ++r) {
            int   j   = jbase + r + (half ? 8 : 0);
            float sj  = slds[j];
            float sj1 = slds[j + 1];
            float wc  = cw[r];
            float wi  = cwi[r];
            float dec = fmaxf(wc - 1.0f, 1e-10f);   // clip(w_cum - THRESH, 1e-10, .)
            bool ok = (wc > 1.0f) && (wi >= sj * dec) && (wi < sj1 * dec);
            // min of positive fractions wi/dec without dividing
            bool better = ok && (wi * best_d < best_n * dec);
            best_n = better ? wi  : best_n;
            best_d = better ? dec : best_d;
        }

        // carry = last row (M=15) of the tile: VGPR7, lanes 16..31 (N = lane-16)
        carry_w  = __shfl(cw[7],  16 + n, 32);
        carry_wi = __shfl(cwi[7], 16 + n, 32);
    }

    // combine the two M-halves that share the same neuron N
    {
        float on = __shfl_xor(best_n, 16, 32);
        float od = __shfl_xor(best_d, 16, 32);
        bool take = (on * best_d < best_n * od);
        best_n = take ? on : best_n;
        best_d = take ? od : best_d;
    }

    if (half == 0) {
        int c = cbase + n;
        out[(b * C_OUT + c) * 1024 + l] = best_n / best_d;
    }
}

// ---------------------------------------------------------------------------
extern "C" void kernel_launch(void* const* d_in, const int* in_sizes, int n_in,
                              void* d_out, int out_size, void* d_ws, size_t ws_size,
                              hipStream_t stream) {
    const float* x = (const float*)d_in[0];     // [4,32,32,32]
    const float* w = (const float*)d_in[1];     // [64,288]
    float* out = (float*)d_out;                 // [4,64,32,32]
    float* wT  = (float*)d_ws;                  // [288,64] scratch (72 KB)

    ttfs_weight_transpose<<<(C_OUT * K_FEAT + 255) / 256, 256, 0, stream>>>(w, wT);
    ttfs_conv_kernel<<<4096, 128, 0, stream>>>(x, wT, out);
}